// GromovWassersteinDistance_26053271617931
// MI455X (gfx1250) — compile-verified
//
#include <hip/hip_runtime.h>
#include <hip/hip_bf16.h>
#include <math.h>
#include <stdint.h>

// Gromov-Wasserstein distance, MI455X (gfx1250, wave32, WMMA + TDM + async-LDS).
// n = m = 4096, d = 1024, 50 FW iters x 200 Sinkhorn iters.
// Workspace: 8 matrices of N*N f32 + small vectors (~513 MB).

#define NN 4096
#define DD 1024
#define REG 0.005f
#define FW_ITERS 50
#define SK_ITERS 200

typedef float v2f __attribute__((ext_vector_type(2)));
typedef float v8f __attribute__((ext_vector_type(8)));
typedef unsigned int v4u __attribute__((ext_vector_type(4)));
typedef unsigned int v8u __attribute__((ext_vector_type(8)));

#define TM 128
#define TN 128
#define TK 16
#define AS 17    // A tile LDS row stride (1-dword pad per 16) -> conflict-free
#define BSTR 136 // B tile LDS row stride (8-dword pad per 128)

// ---------------------------------------------------------------------------
// CDNA5 async copy: memory -> LDS, tracked by ASYNCcnt.
// ---------------------------------------------------------------------------
__device__ __forceinline__ void async_b128(unsigned lds_off, const float* g) {
  asm volatile("global_load_async_to_lds_b128 %0, %1, off"
               :: "v"(lds_off), "v"(g)
               : "memory");
}
__device__ __forceinline__ void wait_async0() {
  asm volatile("s_wait_asynccnt 0x0" ::: "memory");
}

// ---------------------------------------------------------------------------
// CDNA5 Tensor Data Mover: one descriptor moves a whole 2D tile into LDS,
// hardware-padding rows so the LDS image matches our bank-conflict layout.
// D# group0: count=1 | lds_addr | global_addr[56:0] | type=2   (ISA 8.3)
// D# group1: data_size/pads | tensor dims | tile dims | strides (ISA 8.4)
// Issued by one wave; tracked with TENSORcnt (s_wait_tensorcnt).
// ---------------------------------------------------------------------------
__device__ __forceinline__ void tdm_load_2d(unsigned lds_addr, const void* gptr,
                                            v8u g1) {
  uint64_t ga = (uint64_t)(uintptr_t)gptr;
  v4u g0;
  g0[0] = 1u;                                                  // count = 1
  g0[1] = __builtin_amdgcn_readfirstlane(lds_addr);            // lds_addr
  g0[2] = __builtin_amdgcn_readfirstlane((unsigned)ga);        // global_addr lo
  g0[3] = (__builtin_amdgcn_readfirstlane((unsigned)(ga >> 32)) & 0x01FFFFFFu)
          | (2u << 30);                                        // addr hi | type=2
  asm volatile("tensor_load_to_lds %0, %1" :: "s"(g0), "s"(g1) : "memory");
}

// ---------------------------------------------------------------------------
// NN GEMM, TDM staging (C = A[4096xK] @ B[Kx4096], f32 WMMA 16x16x4,
// double-buffered LDS, one barrier per K-chunk, wave 0 drives the TDM).
// ---------------------------------------------------------------------------
__global__ __launch_bounds__(256) void k_gemm_nn_tdm(
    const float* __restrict__ A, const float* __restrict__ B,
    float* __restrict__ C, int K)
{
  __shared__ float At[2][TM * AS];
  __shared__ float Bt[2][TK * BSTR];

  const int tid  = threadIdx.x;
  const int lane = tid & 31;
  const int wave = tid >> 5;
  const int half = lane >> 4;
  const int lrow = lane & 15;
  const int wm   = wave & 3;
  const int wn   = wave >> 2;
  const int bm   = blockIdx.y * TM;
  const int bn   = blockIdx.x * TN;

  const unsigned atBase = (unsigned)(uintptr_t)&At[0][0];
  const unsigned btBase = (unsigned)(uintptr_t)&Bt[0][0];
  const unsigned aStride = TM * AS * 4;
  const unsigned bStride = TK * BSTR * 4;

  // Loop-invariant descriptor group-1 words (uniform -> SGPRs).
  const unsigned Ks = __builtin_amdgcn_readfirstlane((unsigned)K);
  v8u gA1, gB1;
  // A tile: tile 16 x 128 rows, tensor [4096 x K] stride K, pad 1 per 16 dwords
  gA1[0] = (2u << 16) | (1u << 20) | (3u << 22) | (0u << 25);
  gA1[1] = (Ks & 0xffffu) << 16;            // tensor_dim0 = K (lo)
  gA1[2] = (Ks >> 16) | (0x1000u << 16);    // dim0 hi | tensor_dim1 = 4096 (lo)
  gA1[3] = (16u << 16);                     // dim1 hi | tile_dim0 = 16
  gA1[4] = 128u;                            // tile_dim1 = 128, tile_dim2 = 0
  gA1[5] = Ks;                              // tensor_dim0_stride = K
  gA1[6] = 0u; gA1[7] = 0u;
  // B tile: tile 128 x 16 rows, tensor [K x 4096] stride 4096, pad 8 per 128
  gB1[0] = (2u << 16) | (1u << 20) | (6u << 22) | (7u << 25);
  gB1[1] = 0x1000u << 16;                   // tensor_dim0 = 4096 (lo)
  gB1[2] = (Ks & 0xffffu) << 16;            // dim0 hi | tensor_dim1 = K (lo)
  gB1[3] = (Ks >> 16) | (128u << 16);       // dim1 hi | tile_dim0 = 128
  gB1[4] = 16u;                             // tile_dim1 = 16
  gB1[5] = 4096u;                           // tensor_dim0_stride = 4096
  gB1[6] = 0u; gB1[7] = 0u;

  auto issueTDM = [&](int buf, int k0) {
    tdm_load_2d(atBase + (unsigned)buf * aStride, &A[(size_t)bm * K + k0], gA1);
    tdm_load_2d(btBase + (unsigned)buf * bStride, &B[(size_t)k0 * NN + bn], gB1);
  };

  v8f zero = {};
  v8f acc[2][4];
  for (int i = 0; i < 2; ++i)
    for (int j = 0; j < 4; ++j) acc[i][j] = zero;

  const int nk = K / TK;
  if (wave == 0) issueTDM(0, 0);

  for (int kc = 0; kc < nk; ++kc) {
    const int buf = kc & 1;
    if (wave == 0) __builtin_amdgcn_s_wait_tensorcnt(0);  // tiles landed
    __syncthreads();                                      // publish to all waves
    if (kc + 1 < nk && wave == 0) issueTDM(1 - buf, (kc + 1) * TK);

    const float* Ab = At[buf];
    const float* Bb = Bt[buf];
#pragma unroll
    for (int ks = 0; ks < 4; ++ks) {
      const int kk = ks * 4 + half * 2;
      v2f afrag[2];
      v2f bfrag[4];
#pragma unroll
      for (int mi = 0; mi < 2; ++mi) {
        int row = wm * 32 + mi * 16 + lrow;
        afrag[mi][0] = Ab[row * AS + kk + 0];
        afrag[mi][1] = Ab[row * AS + kk + 1];
      }
#pragma unroll
      for (int ni = 0; ni < 4; ++ni) {
        int col = wn * 64 + ni * 16 + lrow;
        bfrag[ni][0] = Bb[(kk + 0) * BSTR + col];
        bfrag[ni][1] = Bb[(kk + 1) * BSTR + col];
      }
#pragma unroll
      for (int mi = 0; mi < 2; ++mi)
#pragma unroll
        for (int ni = 0; ni < 4; ++ni)
          acc[mi][ni] = __builtin_amdgcn_wmma_f32_16x16x4_f32(
              false, afrag[mi], false, bfrag[ni], (short)0, acc[mi][ni],
              false, false);
    }
  }

#pragma unroll
  for (int mi = 0; mi < 2; ++mi)
#pragma unroll
    for (int ni = 0; ni < 4; ++ni) {
      int col = bn + wn * 64 + ni * 16 + lrow;
#pragma unroll
      for (int vv = 0; vv < 8; ++vv) {
        int row = bm + wm * 32 + mi * 16 + half * 8 + vv;
        C[(size_t)row * NN + col] = acc[mi][ni][vv];
      }
    }
}

// ---------------------------------------------------------------------------
// NN GEMM, async-to-LDS staging (same math; keeps the ASYNCcnt path live).
// ---------------------------------------------------------------------------
__global__ __launch_bounds__(256) void k_gemm_nn_async(
    const float* __restrict__ A, const float* __restrict__ B,
    float* __restrict__ C, int K)
{
  __shared__ float At[2][TM * AS];
  __shared__ float Bt[2][TK * BSTR];

  const int tid  = threadIdx.x;
  const int lane = tid & 31;
  const int wave = tid >> 5;
  const int half = lane >> 4;
  const int lrow = lane & 15;
  const int wm   = wave & 3;
  const int wn   = wave >> 2;
  const int bm   = blockIdx.y * TM;
  const int bn   = blockIdx.x * TN;

  const unsigned atBase = (unsigned)(uintptr_t)&At[0][0];
  const unsigned btBase = (unsigned)(uintptr_t)&Bt[0][0];
  const unsigned aStride = TM * AS * 4;
  const unsigned bStride = TK * BSTR * 4;

  auto issueTiles = [&](int buf, int k0) {
#pragma unroll
    for (int it = 0; it < 2; ++it) {
      int idx = tid + it * 256;
      int row = idx >> 2;
      int kq  = (idx & 3) << 2;
      async_b128(atBase + (unsigned)buf * aStride + (unsigned)(row * AS + kq) * 4,
                 &A[(size_t)(bm + row) * K + k0 + kq]);
    }
#pragma unroll
    for (int it = 0; it < 2; ++it) {
      int idx = tid + it * 256;
      int kk  = idx >> 5;
      int nq  = (idx & 31) << 2;
      async_b128(btBase + (unsigned)buf * bStride + (unsigned)(kk * BSTR + nq) * 4,
                 &B[(size_t)(k0 + kk) * NN + bn + nq]);
    }
  };

  v8f zero = {};
  v8f acc[2][4];
  for (int i = 0; i < 2; ++i)
    for (int j = 0; j < 4; ++j) acc[i][j] = zero;

  const int nk = K / TK;
  issueTiles(0, 0);

  for (int kc = 0; kc < nk; ++kc) {
    const int buf = kc & 1;
    wait_async0();
    __syncthreads();
    if (kc + 1 < nk) issueTiles(1 - buf, (kc + 1) * TK);

    const float* Ab = At[buf];
    const float* Bb = Bt[buf];
#pragma unroll
    for (int ks = 0; ks < 4; ++ks) {
      const int kk = ks * 4 + half * 2;
      v2f afrag[2];
      v2f bfrag[4];
#pragma unroll
      for (int mi = 0; mi < 2; ++mi) {
        int row = wm * 32 + mi * 16 + lrow;
        afrag[mi][0] = Ab[row * AS + kk + 0];
        afrag[mi][1] = Ab[row * AS + kk + 1];
      }
#pragma unroll
      for (int ni = 0; ni < 4; ++ni) {
        int col = wn * 64 + ni * 16 + lrow;
        bfrag[ni][0] = Bb[(kk + 0) * BSTR + col];
        bfrag[ni][1] = Bb[(kk + 1) * BSTR + col];
      }
#pragma unroll
      for (int mi = 0; mi < 2; ++mi)
#pragma unroll
        for (int ni = 0; ni < 4; ++ni)
          acc[mi][ni] = __builtin_amdgcn_wmma_f32_16x16x4_f32(
              false, afrag[mi], false, bfrag[ni], (short)0, acc[mi][ni],
              false, false);
    }
  }

#pragma unroll
  for (int mi = 0; mi < 2; ++mi)
#pragma unroll
    for (int ni = 0; ni < 4; ++ni) {
      int col = bn + wn * 64 + ni * 16 + lrow;
#pragma unroll
      for (int vv = 0; vv < 8; ++vv) {
        int row = bm + wm * 32 + mi * 16 + half * 8 + vv;
        C[(size_t)row * NN + col] = acc[mi][ni][vv];
      }
    }
}

// ---------------------------------------------------------------------------
// NT GEMM (Gram matrices only, K=1024): C = A @ B^T, synchronous staging.
// ---------------------------------------------------------------------------
__global__ __launch_bounds__(256) void k_gemm_nt(
    const float* __restrict__ A, const float* __restrict__ B,
    float* __restrict__ C, int K)
{
  __shared__ float At[TM * AS];
  __shared__ float Bt[TK * BSTR];

  const int tid  = threadIdx.x;
  const int lane = tid & 31;
  const int wave = tid >> 5;
  const int half = lane >> 4;
  const int lrow = lane & 15;
  const int wm   = wave & 3;
  const int wn   = wave >> 2;
  const int bm   = blockIdx.y * TM;
  const int bn   = blockIdx.x * TN;

  v8f zero = {};
  v8f acc[2][4];
  for (int i = 0; i < 2; ++i)
    for (int j = 0; j < 4; ++j) acc[i][j] = zero;

  for (int k0 = 0; k0 < K; k0 += TK) {
    __syncthreads();
#pragma unroll
    for (int it = 0; it < 2; ++it) {
      int idx = tid + it * 256;
      int row = idx >> 2;
      int kq  = (idx & 3) << 2;
      float4 vA = *reinterpret_cast<const float4*>(
          &A[(size_t)(bm + row) * K + k0 + kq]);
      At[row * AS + kq + 0] = vA.x;
      At[row * AS + kq + 1] = vA.y;
      At[row * AS + kq + 2] = vA.z;
      At[row * AS + kq + 3] = vA.w;
    }
#pragma unroll
    for (int it = 0; it < 2; ++it) {
      int idx = tid + it * 256;
      int nn  = idx >> 2;
      int kq  = (idx & 3) << 2;
      float4 vB = *reinterpret_cast<const float4*>(
          &B[(size_t)(bn + nn) * K + k0 + kq]);
      Bt[(kq + 0) * BSTR + nn] = vB.x;
      Bt[(kq + 1) * BSTR + nn] = vB.y;
      Bt[(kq + 2) * BSTR + nn] = vB.z;
      Bt[(kq + 3) * BSTR + nn] = vB.w;
    }
    __syncthreads();

#pragma unroll
    for (int ks = 0; ks < 4; ++ks) {
      const int kk = ks * 4 + half * 2;
      v2f afrag[2];
      v2f bfrag[4];
#pragma unroll
      for (int mi = 0; mi < 2; ++mi) {
        int row = wm * 32 + mi * 16 + lrow;
        afrag[mi][0] = At[row * AS + kk + 0];
        afrag[mi][1] = At[row * AS + kk + 1];
      }
#pragma unroll
      for (int ni = 0; ni < 4; ++ni) {
        int col = wn * 64 + ni * 16 + lrow;
        bfrag[ni][0] = Bt[(kk + 0) * BSTR + col];
        bfrag[ni][1] = Bt[(kk + 1) * BSTR + col];
      }
#pragma unroll
      for (int mi = 0; mi < 2; ++mi)
#pragma unroll
        for (int ni = 0; ni < 4; ++ni)
          acc[mi][ni] = __builtin_amdgcn_wmma_f32_16x16x4_f32(
              false, afrag[mi], false, bfrag[ni], (short)0, acc[mi][ni],
              false, false);
    }
  }

#pragma unroll
  for (int mi = 0; mi < 2; ++mi)
#pragma unroll
    for (int ni = 0; ni < 4; ++ni) {
      int col = bn + wn * 64 + ni * 16 + lrow;
#pragma unroll
      for (int vv = 0; vv < 8; ++vv) {
        int row = bm + wm * 32 + mi * 16 + half * 8 + vv;
        C[(size_t)row * NN + col] = acc[mi][ni][vv];
      }
    }
}

// ---------------------------------------------------------------------------
// Small / streaming kernels
// ---------------------------------------------------------------------------
__global__ void k_fill(float* p, float v, int n) {
  int i = blockIdx.x * blockDim.x + threadIdx.x;
  if (i < n) p[i] = v;
}

__global__ void k_diag(const float* __restrict__ g, float* __restrict__ sq) {
  int i = blockIdx.x * blockDim.x + threadIdx.x;
  if (i < NN) sq[i] = g[(size_t)i * NN + i];
}

__global__ void k_dist(const float* __restrict__ gram, const float* __restrict__ sq,
                       float* __restrict__ Cm) {
  size_t total = (size_t)NN * NN;
  for (size_t idx = (size_t)blockIdx.x * 256 + threadIdx.x; idx < total;
       idx += (size_t)gridDim.x * 256) {
    int i = (int)(idx >> 12), j = (int)(idx & (NN - 1));
    float d2 = sq[i] + sq[j] - 2.f * gram[idx];
    Cm[idx] = sqrtf(fmaxf(d2, 0.f));
  }
}

__global__ __launch_bounds__(256) void k_row_moments(
    const float* __restrict__ Cm, float* __restrict__ u, float* __restrict__ p)
{
  __shared__ float s1[256], s2[256];
  int r = blockIdx.x;
  const float* row = Cm + (size_t)r * NN;
  float a = 0.f, b = 0.f;
  for (int j = threadIdx.x; j < NN; j += 256) {
    float x = row[j];
    a += x * x;
    b += x;
  }
  s1[threadIdx.x] = a; s2[threadIdx.x] = b;
  __syncthreads();
  for (int off = 128; off > 0; off >>= 1) {
    if (threadIdx.x < off) {
      s1[threadIdx.x] += s1[threadIdx.x + off];
      s2[threadIdx.x] += s2[threadIdx.x + off];
    }
    __syncthreads();
  }
  if (threadIdx.x == 0) {
    u[r] = s1[0] * (1.f / NN);
    p[r] = s2[0] * (1.f / NN);
  }
}

__global__ void k_init(const float* __restrict__ p, const float* __restrict__ q,
                       float* __restrict__ G, float* __restrict__ CGC,
                       float* __restrict__ CGCt) {
  const float g0 = 1.f / ((float)NN * (float)NN);
  size_t total = (size_t)NN * NN;
  for (size_t idx = (size_t)blockIdx.x * 256 + threadIdx.x; idx < total;
       idx += (size_t)gridDim.x * 256) {
    int i = (int)(idx >> 12), j = (int)(idx & (NN - 1));
    G[idx]    = g0;
    CGC[idx]  = p[i] * q[j];
    CGCt[idx] = p[j] * q[i];
  }
}

__global__ __launch_bounds__(256) void k_minmax1(
    const float* __restrict__ CGC, const float* __restrict__ u,
    const float* __restrict__ v, float* __restrict__ pmin, float* __restrict__ pmax)
{
  __shared__ float smin[256], smax[256];
  float mn = 3.4e38f, mx = -3.4e38f;
  size_t total = (size_t)NN * NN;
  for (size_t idx = (size_t)blockIdx.x * 256 + threadIdx.x; idx < total;
       idx += (size_t)gridDim.x * 256) {
    int i = (int)(idx >> 12), j = (int)(idx & (NN - 1));
    float e = u[i] + v[j] - 2.f * CGC[idx];
    mn = fminf(mn, e);
    mx = fmaxf(mx, e);
  }
  smin[threadIdx.x] = mn; smax[threadIdx.x] = mx;
  __syncthreads();
  for (int off = 128; off > 0; off >>= 1) {
    if (threadIdx.x < off) {
      smin[threadIdx.x] = fminf(smin[threadIdx.x], smin[threadIdx.x + off]);
      smax[threadIdx.x] = fmaxf(smax[threadIdx.x], smax[threadIdx.x + off]);
    }
    __syncthreads();
  }
  if (threadIdx.x == 0) { pmin[blockIdx.x] = smin[0]; pmax[blockIdx.x] = smax[0]; }
}

__global__ __launch_bounds__(256) void k_minmax2(
    const float* __restrict__ pmin, const float* __restrict__ pmax, float* __restrict__ S)
{
  __shared__ float smin[256], smax[256];
  float mn = 3.4e38f, mx = -3.4e38f;
  for (int i = threadIdx.x; i < 1024; i += 256) {
    mn = fminf(mn, pmin[i]);
    mx = fmaxf(mx, pmax[i]);
  }
  smin[threadIdx.x] = mn; smax[threadIdx.x] = mx;
  __syncthreads();
  for (int off = 128; off > 0; off >>= 1) {
    if (threadIdx.x < off) {
      smin[threadIdx.x] = fminf(smin[threadIdx.x], smin[threadIdx.x + off]);
      smax[threadIdx.x] = fmaxf(smax[threadIdx.x], smax[threadIdx.x + off]);
    }
    __syncthreads();
  }
  if (threadIdx.x == 0) { S[0] = smin[0]; S[1] = smax[0]; }
}

__global__ __launch_bounds__(256) void k_lse(
    const float* __restrict__ Mtx, const float* __restrict__ rv,
    const float* __restrict__ cv, const float* __restrict__ add,
    const float* __restrict__ S, float logw, float* __restrict__ out)
{
  __shared__ float sm[256], ss[256];
  const int r = blockIdx.x;
  const float emin = S[0], emax = S[1];
  const float Dd  = (2.f * (emax - emin) + 1e-12f) * REG;
  const float kA  = 4.f / Dd;
  const float kRC = -2.f / Dd;
  const float rterm = kRC * rv[r] + 2.f * emin / Dd;
  const float* row = Mtx + (size_t)r * NN;

  float m = -3.4e38f, s = 0.f;
  for (int j = threadIdx.x; j < NN; j += 256) {
    __builtin_prefetch(&row[j + 2048], 0, 1);
    float val = kA * row[j] + kRC * cv[j] + add[j] + rterm;
    if (val > m) { s = s * __expf(m - val) + 1.f; m = val; }
    else         { s += __expf(val - m); }
  }
  sm[threadIdx.x] = m; ss[threadIdx.x] = s;
  __syncthreads();
  for (int off = 128; off > 0; off >>= 1) {
    if (threadIdx.x < off) {
      float m1 = sm[threadIdx.x], m2 = sm[threadIdx.x + off];
      float M = fmaxf(m1, m2);
      ss[threadIdx.x] = ss[threadIdx.x] * __expf(m1 - M)
                      + ss[threadIdx.x + off] * __expf(m2 - M);
      sm[threadIdx.x] = M;
    }
    __syncthreads();
  }
  if (threadIdx.x == 0) out[r] = logw - (sm[0] + __logf(ss[0]));
}

__global__ void k_plan(const float* __restrict__ CGC, const float* __restrict__ u,
                       const float* __restrict__ v, const float* __restrict__ f,
                       const float* __restrict__ g, const float* __restrict__ S,
                       float* __restrict__ Gc)
{
  const float emin = S[0], emax = S[1];
  const float Dd  = (2.f * (emax - emin) + 1e-12f) * REG;
  const float kA  = 4.f / Dd;
  const float kRC = -2.f / Dd;
  const float kD  = 2.f * emin / Dd;
  size_t total = (size_t)NN * NN;
  for (size_t idx = (size_t)blockIdx.x * 256 + threadIdx.x; idx < total;
       idx += (size_t)gridDim.x * 256) {
    int i = (int)(idx >> 12), j = (int)(idx & (NN - 1));
    float lk = kA * CGC[idx] + kRC * (u[i] + v[j]) + kD;
    Gc[idx] = __expf(f[i] + lk + g[j]);
  }
}

__global__ __launch_bounds__(256) void k_lsred(
    const float* __restrict__ CGC, const float* __restrict__ CGCc,
    const float* __restrict__ G, const float* __restrict__ Gc,
    const float* __restrict__ u, const float* __restrict__ v, float* __restrict__ S)
{
  __shared__ float sh1[256], sh2[256], sh3[256];
  float t1 = 0.f, t2 = 0.f, t3 = 0.f;
  size_t total = (size_t)NN * NN;
  for (size_t idx = (size_t)blockIdx.x * 256 + threadIdx.x; idx < total;
       idx += (size_t)gridDim.x * 256) {
    int i = (int)(idx >> 12), j = (int)(idx & (NN - 1));
    float dd = Gc[idx] - G[idx];
    t1 += (CGCc[idx] - CGC[idx]) * dd;
    t2 += CGC[idx] * dd;
    t3 += (u[i] + v[j]) * dd;
  }
  sh1[threadIdx.x] = t1; sh2[threadIdx.x] = t2; sh3[threadIdx.x] = t3;
  __syncthreads();
  for (int off = 128; off > 0; off >>= 1) {
    if (threadIdx.x < off) {
      sh1[threadIdx.x] += sh1[threadIdx.x + off];
      sh2[threadIdx.x] += sh2[threadIdx.x + off];
      sh3[threadIdx.x] += sh3[threadIdx.x + off];
    }
    __syncthreads();
  }
  if (threadIdx.x == 0) {
    atomicAdd(&S[2], sh1[0]);
    atomicAdd(&S[3], sh2[0]);
    atomicAdd(&S[4], sh3[0]);
  }
}

__global__ void k_alpha(float* S) {
  if (threadIdx.x == 0 && blockIdx.x == 0) {
    float ca = -2.f * S[2];
    float cb = S[4] - 4.f * S[3];
    float alpha;
    if (ca > 0.f) alpha = fminf(1.f, fmaxf(0.f, -cb / (2.f * ca)));
    else          alpha = (ca + cb < 0.f) ? 1.f : 0.f;
    S[5] = alpha;
  }
}

__global__ void k_update(float* __restrict__ G, float* __restrict__ CGC,
                         const float* __restrict__ Gc, const float* __restrict__ CGCc,
                         const float* __restrict__ S)
{
  const float al = S[5];
  size_t total = (size_t)NN * NN;
  for (size_t idx = (size_t)blockIdx.x * 256 + threadIdx.x; idx < total;
       idx += (size_t)gridDim.x * 256) {
    G[idx]   += al * (Gc[idx]   - G[idx]);
    CGC[idx] += al * (CGCc[idx] - CGC[idx]);
  }
}

__global__ void k_transpose(const float* __restrict__ src, float* __restrict__ dst) {
  __shared__ float t[32][33];
  int bx = blockIdx.x * 32, by = blockIdx.y * 32;
  int x = bx + threadIdx.x;
  for (int r = threadIdx.y; r < 32; r += 8)
    t[r][threadIdx.x] = src[(size_t)(by + r) * NN + x];
  __syncthreads();
  int x2 = by + threadIdx.x;
  for (int r = threadIdx.y; r < 32; r += 8)
    dst[(size_t)(bx + r) * NN + x2] = t[threadIdx.x][r];
}

__global__ __launch_bounds__(256) void k_loss(
    const float* __restrict__ CGC, const float* __restrict__ G,
    const float* __restrict__ u, const float* __restrict__ v, float* __restrict__ S)
{
  __shared__ float sh[256];
  float t = 0.f;
  size_t total = (size_t)NN * NN;
  for (size_t idx = (size_t)blockIdx.x * 256 + threadIdx.x; idx < total;
       idx += (size_t)gridDim.x * 256) {
    int i = (int)(idx >> 12), j = (int)(idx & (NN - 1));
    t += (u[i] + v[j] - 2.f * CGC[idx]) * G[idx];
  }
  sh[threadIdx.x] = t;
  __syncthreads();
  for (int off = 128; off > 0; off >>= 1) {
    if (threadIdx.x < off) sh[threadIdx.x] += sh[threadIdx.x + off];
    __syncthreads();
  }
  if (threadIdx.x == 0) atomicAdd(&S[6], sh[0]);
}

__global__ void k_out(const float* __restrict__ S, float* __restrict__ out) {
  if (threadIdx.x == 0 && blockIdx.x == 0) out[0] = S[6];
}

// ---------------------------------------------------------------------------
extern "C" void kernel_launch(void* const* d_in, const int* in_sizes, int n_in,
                              void* d_out, int out_size, void* d_ws, size_t ws_size,
                              hipStream_t stream)
{
  (void)in_sizes; (void)n_in; (void)out_size; (void)ws_size;
  const float* x = (const float*)d_in[0];   // [4096, 1024]
  const float* y = (const float*)d_in[1];   // [4096, 1024]
  // d_in[2] (labels) is unused by the reference output.

  float* w = (float*)d_ws;
  const size_t MAT = (size_t)NN * NN;
  float* C1   = w + 0 * MAT;
  float* C2   = w + 1 * MAT;
  float* G    = w + 2 * MAT;
  float* CGC  = w + 3 * MAT;
  float* CGCt = w + 4 * MAT;
  float* Gc   = w + 5 * MAT;
  float* CGCc = w + 6 * MAT;
  float* TMP  = w + 7 * MAT;
  float* vu = w + 8 * MAT;
  float* vv = vu + NN;
  float* vp = vv + NN;
  float* vq = vp + NN;
  float* vf = vq + NN;
  float* vg = vf + NN;
  float* S  = vg + NN;      // [0]emin [1]emax [2..4]ls sums [5]alpha [6]loss
  float* pmin = S + 16;     // 1024
  float* pmax = pmin + 1024;

  const dim3 gemmGrid(NN / TN, NN / TM);
  const dim3 trGrid(NN / 32, NN / 32), trBlk(32, 8);
  const int EW = 4096;
  const float loga = logf(1.f / NN);

  // ---- distance matrices (Gram via f32 WMMA NT, then sqrt form)
  k_gemm_nt<<<gemmGrid, 256, 0, stream>>>(x, x, TMP, DD);
  k_diag<<<NN / 256, 256, 0, stream>>>(TMP, vp);
  k_dist<<<EW, 256, 0, stream>>>(TMP, vp, C1);
  k_gemm_nt<<<gemmGrid, 256, 0, stream>>>(y, y, TMP, DD);
  k_diag<<<NN / 256, 256, 0, stream>>>(TMP, vp);
  k_dist<<<EW, 256, 0, stream>>>(TMP, vp, C2);

  // ---- constC rank-2 factors and rank-1 initial CGC
  k_row_moments<<<NN, 256, 0, stream>>>(C1, vu, vp);
  k_row_moments<<<NN, 256, 0, stream>>>(C2, vv, vq);
  k_init<<<EW, 256, 0, stream>>>(vp, vq, G, CGC, CGCt);

  // ---- Frank-Wolfe
  for (int it = 0; it < FW_ITERS; ++it) {
    k_minmax1<<<1024, 256, 0, stream>>>(CGC, vu, vv, pmin, pmax);
    k_minmax2<<<1, 256, 0, stream>>>(pmin, pmax, S);
    k_fill<<<NN / 256, 256, 0, stream>>>(vf, 0.f, NN);
    k_fill<<<NN / 256, 256, 0, stream>>>(vg, 0.f, NN);
    for (int sk = 0; sk < SK_ITERS; ++sk) {
      k_lse<<<NN, 256, 0, stream>>>(CGC,  vu, vv, vg, S, loga, vf);
      k_lse<<<NN, 256, 0, stream>>>(CGCt, vv, vu, vf, S, loga, vg);
    }
    k_plan<<<EW, 256, 0, stream>>>(CGC, vu, vv, vf, vg, S, Gc);
    // CGCc = C1 @ Gc @ C2: TDM staging for the first, async-LDS for the second.
    k_gemm_nn_tdm<<<gemmGrid, 256, 0, stream>>>(C1, Gc, TMP, NN);
    k_gemm_nn_async<<<gemmGrid, 256, 0, stream>>>(TMP, C2, CGCc, NN);
    k_fill<<<1, 16, 0, stream>>>(S + 2, 0.f, 3);
    k_lsred<<<1024, 256, 0, stream>>>(CGC, CGCc, G, Gc, vu, vv, S);
    k_alpha<<<1, 32, 0, stream>>>(S);
    k_update<<<EW, 256, 0, stream>>>(G, CGC, Gc, CGCc, S);
    k_transpose<<<trGrid, trBlk, 0, stream>>>(CGC, CGCt);
  }

  // ---- final gw loss
  k_fill<<<1, 16, 0, stream>>>(S + 6, 0.f, 1);
  k_loss<<<1024, 256, 0, stream>>>(CGC, G, vu, vv, S);
  k_out<<<1, 32, 0, stream>>>(S, (float*)d_out);
}